// PLS_91311004713195
// MI455X (gfx1250) — compile-verified
//
#include <hip/hip_runtime.h>

typedef __attribute__((ext_vector_type(2))) float v2f;
typedef __attribute__((ext_vector_type(4))) float f4;
typedef __attribute__((ext_vector_type(8))) float v8f;

#define NMAT 2048
#define DDIM 512
#define MPL  8
#define NELEM (NMAT * NMAT)       // 4M elements per plane
#define NV4   (NELEM / 4)         // 1M float4 per plane

// ws layout (floats): [0..7]=scores, [8..43]=G upper triangle, [44..51]=coef
#define WS_NRED 44
#define WS_COEF 44
#define WS_TOTAL 52

// ---------------------------------------------------------------------------
// Kernel 0: zero the reduction workspace (graph-replay safe, re-zeroed each call)
// ---------------------------------------------------------------------------
__global__ void PLS_zero_ws(float* ws) {
    int i = threadIdx.x;
    if (i < WS_TOTAL) ws[i] = 0.0f;
}

// ---------------------------------------------------------------------------
// Kernel 1: Y0 = X X^T via V_WMMA_F32_16X16X4_F32 (exact f32 math).
// One wave per 16x16 output tile; 8 waves (256 threads) per block.
// A-tile layout (f32 16x4): lane 0-15 -> M=lane, VGPR0=K0,VGPR1=K1;
//                           lane16-31 -> M=lane-16, VGPR0=K2,VGPR1=K3.
// B-tile mirrors with N across lanes. Both tiles read from X -> same formula.
// ---------------------------------------------------------------------------
__global__ __launch_bounds__(256) void PLS_gemm_xxt(const float* __restrict__ X,
                                                    float* __restrict__ Y) {
    const int lane = threadIdx.x & 31;
    const int wid  = threadIdx.x >> 5;
    const int half = lane >> 4;        // 0 or 1
    const int l16  = lane & 15;

    const int tile = blockIdx.x * 8 + wid;     // 16384 tiles (128 x 128)
    const int tr = tile >> 7;
    const int tc = tile & 127;
    const int m0 = tr * 16;
    const int n0 = tc * 16;

    const float* __restrict__ xa = X + (size_t)(m0 + l16) * DDIM + 2 * half;
    const float* __restrict__ xb = X + (size_t)(n0 + l16) * DDIM + 2 * half;

    v8f acc = {};
    #pragma unroll 4
    for (int kb = 0; kb < DDIM; kb += 4) {
        v2f a = *(const v2f*)(xa + kb);
        v2f b = *(const v2f*)(xb + kb);
        acc = __builtin_amdgcn_wmma_f32_16x16x4_f32(
            /*neg_a=*/false, a, /*neg_b=*/false, b,
            /*c_mod=*/(short)0, acc, /*reuse_a=*/false, /*reuse_b=*/false);
    }

    // C/D layout: VGPR r holds row (8*half + r), N = l16.
    float* yr = Y + (size_t)(m0 + 8 * half) * NMAT + n0 + l16;
    #pragma unroll
    for (int r = 0; r < 8; ++r) yr[(size_t)r * NMAT] = acc[r];
}

// ---------------------------------------------------------------------------
// Kernel 2: single fused pass over A (128MB) + Y0 (16MB):
//   scores[m] = sum A_m .* Y0      (8 accumulators)
//   G[m][t]   = sum A_m .* A_t     (36 upper-triangular accumulators)
// wave32 shfl_xor reduce -> LDS cross-wave -> one f32 atomic per block/acc.
// ---------------------------------------------------------------------------
__global__ __launch_bounds__(256) void PLS_reduce_sg(const float* __restrict__ A,
                                                     const float* __restrict__ Y,
                                                     float* __restrict__ ws) {
    float acc[WS_NRED];
    #pragma unroll
    for (int i = 0; i < WS_NRED; ++i) acc[i] = 0.0f;

    const f4* __restrict__ Y4 = (const f4*)Y;
    const f4* __restrict__ A4 = (const f4*)A;

    const int stride = gridDim.x * blockDim.x;
    for (int i = blockIdx.x * blockDim.x + threadIdx.x; i < NV4; i += stride) {
        f4 y = Y4[i];
        f4 a[MPL];
        #pragma unroll
        for (int m = 0; m < MPL; ++m) a[m] = A4[(size_t)m * NV4 + i];
        #pragma unroll
        for (int c = 0; c < 4; ++c) {
            float yc = y[c];
            float av[MPL];
            #pragma unroll
            for (int m = 0; m < MPL; ++m) av[m] = a[m][c];
            #pragma unroll
            for (int m = 0; m < MPL; ++m) acc[m] += av[m] * yc;
            int u = MPL;
            #pragma unroll
            for (int m = 0; m < MPL; ++m)
                #pragma unroll
                for (int t = m; t < MPL; ++t) acc[u++] += av[m] * av[t];
        }
    }

    // wave32 reduction
    const int lane = threadIdx.x & 31;
    const int wid  = threadIdx.x >> 5;
    __shared__ float sm[8][WS_NRED];
    #pragma unroll
    for (int i = 0; i < WS_NRED; ++i) {
        float v = acc[i];
        #pragma unroll
        for (int o = 16; o > 0; o >>= 1) v += __shfl_xor(v, o, 32);
        if (lane == 0) sm[wid][i] = v;
    }
    __syncthreads();
    if (threadIdx.x < WS_NRED) {
        float v = 0.0f;
        #pragma unroll
        for (int w = 0; w < 8; ++w) v += sm[w][threadIdx.x];
        atomicAdd(&ws[threadIdx.x], v);
    }
}

// ---------------------------------------------------------------------------
// Kernel 3: 8-dimensional solver (1 thread). Runs the K-iteration recurrence
// entirely in the 8-dim span of A: s' = s - a*g, G' = G - g g^T/den,
// T' = T - beta w^T, coef += alpha * w.  Writes coef[8] to ws[44..51].
// ---------------------------------------------------------------------------
__global__ void PLS_solve(float* __restrict__ ws, const int* __restrict__ Kp) {
    if (threadIdx.x != 0 || blockIdx.x != 0) return;
    float s[MPL], G[MPL][MPL], T[MPL][MPL], coef[MPL];
    #pragma unroll
    for (int m = 0; m < MPL; ++m) { s[m] = ws[m]; coef[m] = 0.0f; }
    int u = MPL;
    for (int m = 0; m < MPL; ++m)
        for (int t = m; t < MPL; ++t) { G[m][t] = ws[u]; G[t][m] = ws[u]; ++u; }
    for (int m = 0; m < MPL; ++m)
        for (int j = 0; j < MPL; ++j) T[m][j] = (m == j) ? 1.0f : 0.0f;

    const int K = Kp[0];
    for (int k = 0; k < K; ++k) {
        float g[MPL], w[MPL];
        for (int m = 0; m < MPL; ++m) {
            float acc = 0.0f;
            for (int j = 0; j < MPL; ++j) acc += G[m][j] * s[j];
            g[m] = acc;
        }
        float num = 0.0f, den = 0.0f;
        for (int m = 0; m < MPL; ++m) { num += s[m] * s[m]; den += s[m] * g[m]; }
        const float alpha = num / den;
        for (int j = 0; j < MPL; ++j) {
            float acc = 0.0f;
            for (int m = 0; m < MPL; ++m) acc += T[m][j] * s[m];
            w[j] = acc;
        }
        for (int j = 0; j < MPL; ++j) coef[j] += alpha * w[j];
        for (int m = 0; m < MPL; ++m) {
            const float beta = g[m] / den;
            for (int j = 0; j < MPL; ++j) T[m][j] -= beta * w[j];
        }
        for (int m = 0; m < MPL; ++m)
            for (int t = 0; t < MPL; ++t) G[m][t] -= g[m] * g[t] / den;
        for (int m = 0; m < MPL; ++m) s[m] -= alpha * g[m];
    }
    #pragma unroll
    for (int j = 0; j < MPL; ++j) ws[WS_COEF + j] = coef[j];
}

// ---------------------------------------------------------------------------
// Kernel 4: Y -= sum_m coef[m] * A_m   (single streaming pass, float4)
// ---------------------------------------------------------------------------
__global__ __launch_bounds__(256) void PLS_update_y(const float* __restrict__ A,
                                                    float* __restrict__ Y,
                                                    const float* __restrict__ ws) {
    float c[MPL];
    #pragma unroll
    for (int m = 0; m < MPL; ++m) c[m] = ws[WS_COEF + m];

    f4* __restrict__ Y4 = (f4*)Y;
    const f4* __restrict__ A4 = (const f4*)A;
    const int stride = gridDim.x * blockDim.x;
    for (int i = blockIdx.x * blockDim.x + threadIdx.x; i < NV4; i += stride) {
        f4 y = Y4[i];
        #pragma unroll
        for (int m = 0; m < MPL; ++m) y -= c[m] * A4[(size_t)m * NV4 + i];
        Y4[i] = y;
    }
}

extern "C" void kernel_launch(void* const* d_in, const int* in_sizes, int n_in,
                              void* d_out, int out_size, void* d_ws, size_t ws_size,
                              hipStream_t stream) {
    const float* X = (const float*)d_in[0];   // [2048, 512]
    const float* A = (const float*)d_in[1];   // [8, 2048, 2048]
    const int*   K = (const int*)d_in[2];     // scalar
    float* Y  = (float*)d_out;                // [2048, 2048]
    float* ws = (float*)d_ws;

    PLS_zero_ws<<<1, 64, 0, stream>>>(ws);
    PLS_gemm_xxt<<<2048, 256, 0, stream>>>(X, Y);        // 16384 tiles / 8 waves
    PLS_reduce_sg<<<512, 256, 0, stream>>>(A, Y, ws);
    PLS_solve<<<1, 1, 0, stream>>>(ws, K);
    PLS_update_y<<<1024, 256, 0, stream>>>(A, Y, ws);
}